// EdgeGAT_85401129714159
// MI455X (gfx1250) — compile-verified
//
#include <hip/hip_runtime.h>
#include <hip/hip_bf16.h>
#include <math.h>

// ---------------------------------------------------------------------------
// EdgeGAT for MI455X (gfx1250, wave32, WMMA).
// N=50000 nodes, E=1.2M edges, F_IN=128, F_EDGE=32, H=78 (padded to 80).
// All GEMMs use full-precision V_WMMA_F32_16X16X4_F32 (fp32 reference).
// ---------------------------------------------------------------------------

#define HP 80          // H padded to 5 x 16 WMMA tiles
#define HREAL 78
#define NTILES (HP / 16)

typedef float v2f __attribute__((ext_vector_type(2)));
typedef float v8f __attribute__((ext_vector_type(8)));

// ---------------- helpers ----------------

__device__ __forceinline__ void atomicMaxF(float* addr, float value) {
  // sign-split trick: needs only int/uint atomics, valid for all IEEE floats.
  if (value >= 0.0f) {
    atomicMax((int*)addr, __float_as_int(value));
  } else {
    atomicMin((unsigned int*)addr, __float_as_uint(value));
  }
}

// ---------------- weight prep ----------------

// Packed-B layout for WMMA f32 16x16x4: B row-pair p = i/2 interleaved so a lane's
// (K=kk, K=kk+1) pair at column j is one contiguous b64:
//   Bq[(p*HP + j)*2 + (i&1)] = B[i][j]   (zero-padded to R x HP)
__global__ void k_pad_packB(const float* __restrict__ src, float* __restrict__ dst,
                            int r, int c, int R) {
  int idx = blockIdx.x * blockDim.x + threadIdx.x;
  if (idx >= R * HP) return;
  int i = idx / HP, j = idx % HP;
  float val = (i < r && j < c) ? src[i * c + j] : 0.0f;
  dst[((size_t)(i >> 1) * HP + j) * 2 + (i & 1)] = val;
}

// v[j] = sum_i Wm2[j][i] * Wf[2H+i]  (folds second edge-MLP GEMM into the head)
// c    = bm2 . Wf3 + bf
__global__ void k_vc(const float* __restrict__ Wm2, const float* __restrict__ bm2,
                     const float* __restrict__ Wf, const float* __restrict__ bf,
                     float* __restrict__ v, float* __restrict__ cbuf) {
  int j = threadIdx.x;
  if (j < HP) {
    float acc = 0.0f;
    if (j < HREAL) {
      for (int i = 0; i < HREAL; ++i) acc += Wm2[j * HREAL + i] * Wf[2 * HREAL + i];
    }
    v[j] = acc;
  }
  if (j == 0) {
    float c = bf[0];
    for (int i = 0; i < HREAL; ++i) c += bm2[i] * Wf[2 * HREAL + i];
    cbuf[0] = c;
  }
}

// ---------------- fp32 WMMA GEMM: C[MxHP] = A[MxK(lda)] @ Bq[packed KxHP] --------
// Row addresses are clamped (not masked): out-of-range rows compute garbage that is
// never stored, keeping EXEC uniform and loads branch-free inside the WMMA loop.

template <int K>
__global__ void k_gemm_wmma(const float* __restrict__ A, const float* __restrict__ Bq,
                            float* __restrict__ C, int M, int lda) {
  int wid  = (blockIdx.x * blockDim.x + threadIdx.x) >> 5;
  int lane = threadIdx.x & 31;
  int nWaves = ((M + 15) >> 4) * NTILES;
  if (wid >= nWaves) return;          // wave-uniform exit: EXEC all-ones at WMMA
  int mtile = wid / NTILES;
  int ntile = wid % NTILES;
  int m0 = mtile << 4, n0 = ntile << 4;
  int half = lane >> 4, l16 = lane & 15;
  int row = m0 + l16;
  int rowA = row < M ? row : M - 1;   // clamp: branch-free
  int col = n0 + l16;

  const float* Ar = A + (size_t)rowA * lda + (half << 1);
  const float* Br = Bq + ((size_t)half * HP + col) * 2;

  v8f acc = {};
#pragma unroll
  for (int k = 0; k < K; k += 4) {
    v2f a = *(const v2f*)(Ar + k);
    v2f b = *(const v2f*)(Br + (size_t)(k >> 1) * HP * 2);
    acc = __builtin_amdgcn_wmma_f32_16x16x4_f32(false, a, false, b,
                                                (short)0, acc, false, false);
  }
#pragma unroll
  for (int i = 0; i < 8; ++i) {
    int r = m0 + i + (half << 3);
    if (r < M) C[(size_t)r * HP + col] = acc[i];
  }
}

// ---------------- attention scalars: alpha_s/d[n] = h[n] . a ----------------

__global__ void k_alphas(const float* __restrict__ h, const float* __restrict__ a_s,
                         const float* __restrict__ a_d, float* __restrict__ os,
                         float* __restrict__ od, int N) {
  int n = blockIdx.x * blockDim.x + threadIdx.x;
  if (n >= N) return;
  float s1 = 0.0f, s2 = 0.0f;
  const float* hr = h + (size_t)n * HP;
  for (int j = 0; j < HREAL; ++j) { float hv = hr[j]; s1 += hv * a_s[j]; s2 += hv * a_d[j]; }
  os[n] = s1; od[n] = s2;
}

// ---------------- aggregation ----------------

__global__ void k_init(float* __restrict__ m, float* __restrict__ s,
                       float* __restrict__ agg, int N) {
  int idx = blockIdx.x * blockDim.x + threadIdx.x;
  if (idx < N * HP) agg[idx] = 0.0f;
  if (idx < N) { m[idx] = -__builtin_inff(); s[idx] = 0.0f; }
}

__global__ void k_edge_max(const int* __restrict__ ei, int E, int N,
                           const float* __restrict__ as_, const float* __restrict__ ad_,
                           float* __restrict__ m) {
  int idx = blockIdx.x * blockDim.x + threadIdx.x;
  int EP = E + N;
  if (idx >= EP) return;
  int src, dst;
  if (idx < E) { src = ei[idx]; dst = ei[E + idx]; } else { src = dst = idx - E; }
  float e = as_[src] + ad_[dst];
  e = e > 0.0f ? e : 0.2f * e;            // leaky_relu(0.2)
  atomicMaxF(&m[dst], e);
}

// one wave per edge: w = exp(e - m[dst]); s[dst]+=w; agg[dst] += w*h[src]
// h/agg (16 MB each) stay L2-resident; atomics resolve in L2.
__global__ void k_edge_scatter(const int* __restrict__ ei, int E, int N,
                               const float* __restrict__ as_, const float* __restrict__ ad_,
                               const float* __restrict__ m, float* __restrict__ s,
                               const float* __restrict__ h, float* __restrict__ agg) {
  int wid  = (blockIdx.x * blockDim.x + threadIdx.x) >> 5;
  int lane = threadIdx.x & 31;
  int EP = E + N;
  if (wid >= EP) return;
  int src, dst;
  if (wid < E) { src = ei[wid]; dst = ei[E + wid]; } else { src = dst = wid - E; }
  float e = as_[src] + ad_[dst];
  e = e > 0.0f ? e : 0.2f * e;
  float w = expf(e - m[dst]);
  if (lane == 0) atomicAdd(&s[dst], w);
  const float* hs = h + (size_t)src * HP;
  float* ag = agg + (size_t)dst * HP;
  for (int j = lane; j < HREAL; j += 32) atomicAdd(&ag[j], w * hs[j]);
}

// out[n][j<78] = (relu?)(agg/(s+eps) + b);  pad cols zeroed (clean K=80 next GEMM)
__global__ void k_finalize(const float* __restrict__ agg, const float* __restrict__ s,
                           const float* __restrict__ b, float* __restrict__ out,
                           int N, int do_relu) {
  int idx = blockIdx.x * blockDim.x + threadIdx.x;
  if (idx >= N * HP) return;
  int n = idx / HP, j = idx % HP;
  float v = 0.0f;
  if (j < HREAL) {
    v = agg[idx] / (s[n] + 1e-16f) + b[j];
    if (do_relu) v = v > 0.0f ? v : 0.0f;
  }
  out[idx] = v;
}

// ---------------- head precompute: p1 = h2 . Wf[0:H], p2 = h2 . Wf[H:2H] -----------

__global__ void k_pscalars(const float* __restrict__ h2, const float* __restrict__ Wf,
                           float* __restrict__ p1, float* __restrict__ p2, int N) {
  int n = blockIdx.x * blockDim.x + threadIdx.x;
  if (n >= N) return;
  float a = 0.0f, b = 0.0f;
  const float* hr = h2 + (size_t)n * HP;
  for (int j = 0; j < HREAL; ++j) { float hv = hr[j]; a += hv * Wf[j]; b += hv * Wf[HREAL + j]; }
  p1[n] = a; p2[n] = b;
}

// ---------------- fused edge MLP + head contraction (WMMA) ----------------
// term[e] = sum_j relu((edge_attr @ Wm1)[e][j] + bm1[j]) * v[j]
// edge_attr is streamed once -> non-temporal loads keep node data in L2.

__global__ void k_edge_term(const float* __restrict__ EA, const float* __restrict__ Wm1q,
                            const float* __restrict__ bm1, const float* __restrict__ v,
                            float* __restrict__ term, int E) {
  int wid  = (blockIdx.x * blockDim.x + threadIdx.x) >> 5;
  int lane = threadIdx.x & 31;
  int nWaves = (E + 15) >> 4;
  if (wid >= nWaves) return;          // wave-uniform
  int m0 = wid << 4;
  int half = lane >> 4, l16 = lane & 15;
  int row = m0 + l16;
  int rowA = row < E ? row : E - 1;   // clamp: branch-free

  const float* Ar = EA + (size_t)rowA * 32 + (half << 1);

  float rs[8];
#pragma unroll
  for (int i = 0; i < 8; ++i) rs[i] = 0.0f;

  for (int nt = 0; nt < NTILES; ++nt) {
    int n0 = nt << 4;
    int col = n0 + l16;
    const float* Br = Wm1q + ((size_t)half * HP + col) * 2;
    v8f acc = {};
#pragma unroll
    for (int k = 0; k < 32; k += 4) {
      v2f a = __builtin_nontemporal_load((const v2f*)(Ar + k));
      v2f b = *(const v2f*)(Br + (size_t)(k >> 1) * HP * 2);
      acc = __builtin_amdgcn_wmma_f32_16x16x4_f32(false, a, false, b,
                                                  (short)0, acc, false, false);
    }
    float bb = (col < HREAL) ? bm1[col] : 0.0f;
    float vv = v[col];                 // padded entries are zero
#pragma unroll
    for (int i = 0; i < 8; ++i) {
      float t = acc[i] + bb;
      t = t > 0.0f ? t : 0.0f;
      rs[i] += t * vv;
    }
  }
  // reduce each row-sum across the 16 lanes of this half-wave
#pragma unroll
  for (int i = 0; i < 8; ++i) {
    float x = rs[i];
    for (int off = 8; off > 0; off >>= 1) x += __shfl_xor(x, off, 16);
    rs[i] = x;
  }
  if (l16 == 0) {
#pragma unroll
    for (int i = 0; i < 8; ++i) {
      int r = m0 + i + (half << 3);
      if (r < E) term[r] = rs[i];
    }
  }
}

// ---------------- final edge output ----------------

__global__ void k_final_out(const int* __restrict__ ei, const float* __restrict__ p1,
                            const float* __restrict__ p2, const float* __restrict__ term,
                            const float* __restrict__ cbuf, float* __restrict__ out, int E) {
  int e = blockIdx.x * blockDim.x + threadIdx.x;
  if (e >= E) return;
  out[e] = p1[ei[e]] + p2[ei[E + e]] + term[e] + cbuf[0];
}

// ---------------------------------------------------------------------------

static inline int ceil_div(long long a, long long b) { return (int)((a + b - 1) / b); }

extern "C" void kernel_launch(void* const* d_in, const int* in_sizes, int n_in,
                              void* d_out, int out_size, void* d_ws, size_t ws_size,
                              hipStream_t stream) {
  const float* x    = (const float*)d_in[0];
  const int*   ei   = (const int*)  d_in[1];
  const float* ea   = (const float*)d_in[2];
  const float* W1   = (const float*)d_in[3];
  const float* as1  = (const float*)d_in[4];
  const float* ad1  = (const float*)d_in[5];
  const float* b1   = (const float*)d_in[6];
  const float* W2   = (const float*)d_in[7];
  const float* as2  = (const float*)d_in[8];
  const float* ad2  = (const float*)d_in[9];
  const float* b2   = (const float*)d_in[10];
  const float* Wm1  = (const float*)d_in[11];
  const float* bm1  = (const float*)d_in[12];
  const float* Wm2  = (const float*)d_in[13];
  const float* bm2  = (const float*)d_in[14];
  const float* Wf   = (const float*)d_in[15];
  const float* bf   = (const float*)d_in[16];
  float* out = (float*)d_out;

  const int N = in_sizes[0] / 128;   // 50000
  const int E = in_sizes[2] / 32;    // 1200000
  const int EP = E + N;

  // workspace layout (floats)
  float* ws = (float*)d_ws;
  size_t o = 0;
  float* W1q  = ws + o; o += 128 * HP;   // packed row-pair layout
  float* W2q  = ws + o; o += HP * HP;
  float* Wm1q = ws + o; o += 32 * HP;
  float* vvec = ws + o; o += HP;
  float* cbuf = ws + o; o += 4;
  float* alS  = ws + o; o += N;
  float* alD  = ws + o; o += N;
  float* mbuf = ws + o; o += N;
  float* sbuf = ws + o; o += N;
  float* p1   = ws + o; o += N;
  float* p2   = ws + o; o += N;
  float* term = ws + o; o += E;
  float* bufA = ws + o; o += (size_t)N * HP;   // h (pre-aggregation features)
  float* bufB = ws + o; o += (size_t)N * HP;   // layer output / h2
  float* agg  = ws + o; o += (size_t)N * HP;
  (void)ws_size; (void)n_in; (void)out_size;

  const int B = 256;
  int gemmWaves = ((N + 15) >> 4) * NTILES;
  int gemmBlocks = ceil_div((long long)gemmWaves * 32, B);
  int nhpBlocks  = ceil_div((long long)N * HP, B);

  // ---- GAT layer 1 ----
  k_pad_packB<<<ceil_div(128 * HP, B), B, 0, stream>>>(W1, W1q, 128, HREAL, 128);
  k_gemm_wmma<128><<<gemmBlocks, B, 0, stream>>>(x, W1q, bufA, N, 128);
  k_alphas<<<ceil_div(N, B), B, 0, stream>>>(bufA, as1, ad1, alS, alD, N);
  k_init<<<nhpBlocks, B, 0, stream>>>(mbuf, sbuf, agg, N);
  k_edge_max<<<ceil_div(EP, B), B, 0, stream>>>(ei, E, N, alS, alD, mbuf);
  k_edge_scatter<<<ceil_div((long long)EP * 32, B), B, 0, stream>>>(ei, E, N, alS, alD,
                                                                    mbuf, sbuf, bufA, agg);
  k_finalize<<<nhpBlocks, B, 0, stream>>>(agg, sbuf, b1, bufB, N, 1);

  // ---- GAT layer 2 ----
  k_pad_packB<<<ceil_div(HP * HP, B), B, 0, stream>>>(W2, W2q, HREAL, HREAL, HP);
  k_gemm_wmma<HP><<<gemmBlocks, B, 0, stream>>>(bufB, W2q, bufA, N, HP);
  k_alphas<<<ceil_div(N, B), B, 0, stream>>>(bufA, as2, ad2, alS, alD, N);
  k_init<<<nhpBlocks, B, 0, stream>>>(mbuf, sbuf, agg, N);
  k_edge_max<<<ceil_div(EP, B), B, 0, stream>>>(ei, E, N, alS, alD, mbuf);
  k_edge_scatter<<<ceil_div((long long)EP * 32, B), B, 0, stream>>>(ei, E, N, alS, alD,
                                                                    mbuf, sbuf, bufA, agg);
  k_finalize<<<nhpBlocks, B, 0, stream>>>(agg, sbuf, b2, bufB, N, 0);

  // ---- head precompute ----
  k_pscalars<<<ceil_div(N, B), B, 0, stream>>>(bufB, Wf, p1, p2, N);
  k_pad_packB<<<ceil_div(32 * HP, B), B, 0, stream>>>(Wm1, Wm1q, 32, HREAL, 32);
  k_vc<<<1, 128, 0, stream>>>(Wm2, bm2, Wf, bf, vvec, cbuf);

  // ---- fused edge MLP + head contraction ----
  int etWaves = (E + 15) >> 4;
  k_edge_term<<<ceil_div((long long)etWaves * 32, B), B, 0, stream>>>(ea, Wm1q, bm1, vvec,
                                                                      term, E);

  // ---- final output ----
  k_final_out<<<ceil_div(E, B), B, 0, stream>>>(ei, p1, p2, term, cbuf, out, E);
}